// PhysicsFormer_79130477461817
// MI455X (gfx1250) — compile-verified
//
#include <hip/hip_runtime.h>
#include <math.h>

// ---------------------------------------------------------------------------
// Geometry
//   reference arrays: (60, 258, 258) f32 after pad1().
//   internal layout : (60, 288, 288), rows/cols [258..288) kept at 0.0 so the
//   288x288x288 GEMM is exact on the valid region with zero pad contribution.
// ---------------------------------------------------------------------------
#define LDIM   288
#define VALIDD 258
#define NCLS   6
#define CHANS  60
#define CH_STRIDE (LDIM * LDIM)   // 82944 floats per channel

#define KB   32                   // K-chunk staged in LDS
#define BPAD 36                   // LDS row pitch (floats): 16B-aligned rows,
                                  // 36*m mod 64 distinct for m=0..15 -> no bank conflicts

typedef float v2f __attribute__((ext_vector_type(2)));
typedef float v8f __attribute__((ext_vector_type(8)));
typedef int   v4i __attribute__((ext_vector_type(4)));

typedef __attribute__((address_space(1))) v4i* gptr_v4i;
typedef __attribute__((address_space(3))) v4i* lptr_v4i;

#if defined(__gfx1250__) && __has_builtin(__builtin_amdgcn_global_load_async_to_lds_b128)
#define HAVE_ASYNC_LDS 1
#else
#define HAVE_ASYNC_LDS 0
#endif

__device__ __forceinline__ void wait_async_lds() {
#if HAVE_ASYNC_LDS
#if __has_builtin(__builtin_amdgcn_s_wait_asynccnt)
  __builtin_amdgcn_s_wait_asynccnt(0);
#else
  asm volatile("s_wait_asynccnt 0" ::: "memory");
#endif
#endif
}

// ------------------------------- init --------------------------------------
__global__ void k_init(float* acc) { acc[0] = 0.f; acc[1] = 0.f; }

// ------------------------- prep: upsample+softmax+pairs+pad ----------------
__global__ void k_prep(const float* __restrict__ logits,
                       float* __restrict__ fcp, float* __restrict__ fcn) {
  int idx = blockIdx.x * blockDim.x + threadIdx.x;
  if (idx >= CH_STRIDE) return;
  int h = idx / LDIM, w = idx % LDIM;
  bool outside = (h >= VALIDD) || (w >= VALIDD);
  bool ring = (!outside) && (h == 0 || h == VALIDD - 1 || w == 0 || w == VALIDD - 1);

  float soft[NCLS];
  if (!outside && !ring) {
    // bilinear upsample 64 -> 256 (half-pixel centers, edge clamp)
    int y = h - 1, x = w - 1;
    float sy = (y + 0.5f) * 0.25f - 0.5f;
    float sx = (x + 0.5f) * 0.25f - 0.5f;
    float fy = floorf(sy), fx = floorf(sx);
    int y0 = (int)fy, x0 = (int)fx;
    float wy = sy - fy, wx = sx - fx;
    int y1 = y0 + 1, x1 = x0 + 1;
    y0 = y0 < 0 ? 0 : (y0 > 63 ? 63 : y0);
    y1 = y1 < 0 ? 0 : (y1 > 63 ? 63 : y1);
    x0 = x0 < 0 ? 0 : (x0 > 63 ? 63 : x0);
    x1 = x1 < 0 ? 0 : (x1 > 63 ? 63 : x1);
    float v[NCLS], m = -1e30f;
    #pragma unroll
    for (int c = 0; c < NCLS; ++c) {
      const float* L = logits + c * 64 * 64;
      float v00 = L[y0 * 64 + x0], v01 = L[y0 * 64 + x1];
      float v10 = L[y1 * 64 + x0], v11 = L[y1 * 64 + x1];
      float val = (1.f - wy) * ((1.f - wx) * v00 + wx * v01) +
                  wy * ((1.f - wx) * v10 + wx * v11);
      v[c] = val;
      m = fmaxf(m, val);
    }
    float s = 0.f;
    #pragma unroll
    for (int c = 0; c < NCLS; ++c) { v[c] = __expf(v[c] - m); s += v[c]; }
    float inv = 1.f / s;
    #pragma unroll
    for (int c = 0; c < NCLS; ++c) soft[c] = v[c] * inv;
  }

  int p = 0;
  #pragma unroll
  for (int i = 0; i < NCLS; ++i) {
    #pragma unroll
    for (int j = 0; j < NCLS; ++j) {
      if (i == j) continue;
      float vp, vn;
      if (outside)      { vp = 0.f; vn = 0.f; }
      else if (ring)    { vp = 1.f; vn = 1.f; }
      else {
        float d = (soft[i] - soft[j]) * 0.5f;
        vp = fmaxf(d, 0.f);
        vn = fmaxf(-d, 0.f);
      }
      size_t o0 = (size_t)(2 * p) * CH_STRIDE + idx;
      size_t o1 = (size_t)(2 * p + 1) * CH_STRIDE + idx;
      fcp[o0] = vp; fcp[o1] = vn;
      if (outside)   { fcn[o0] = 0.f;  fcn[o1] = 0.f; }
      else if (ring) { fcn[o0] = 1.f;  fcn[o1] = 1.f; }
      else           { fcn[o0] = -vp;  fcn[o1] = -vn; }
      ++p;
    }
  }
}

// ------------------------- 3x3 pool (VALID) + re-pad-with-1 ----------------
template <int ISMAX>
__global__ void k_pool(const float* __restrict__ X, float* __restrict__ Y) {
  int idx = blockIdx.x * blockDim.x + threadIdx.x;
  if (idx >= CH_STRIDE) return;
  int ch = blockIdx.y;
  int h = idx / LDIM, w = idx % LDIM;
  const float* x = X + (size_t)ch * CH_STRIDE;
  float out;
  if (h >= VALIDD || w >= VALIDD) {
    out = 0.f;                                   // keep zero pad band
  } else if (h == 0 || h == VALIDD - 1 || w == 0 || w == VALIDD - 1) {
    out = 1.f;                                   // pad1 constant ring
  } else if (ISMAX) {
    float m = -1e30f;
    #pragma unroll
    for (int dy = -1; dy <= 1; ++dy)
      #pragma unroll
      for (int dx = -1; dx <= 1; ++dx)
        m = fmaxf(m, x[(h + dy) * LDIM + (w + dx)]);
    out = m;
  } else {
    float s = 0.f;
    #pragma unroll
    for (int dy = -1; dy <= 1; ++dy)
      #pragma unroll
      for (int dx = -1; dx <= 1; ++dx)
        s += x[(h + dy) * LDIM + (w + dx)];
    out = s * (1.f / 9.f);
  }
  Y[(size_t)ch * CH_STRIDE + idx] = out;
}

// ------------------------- batched 288^3 f32 WMMA GEMM ---------------------
// One wave per 32x32 output macro-tile (2x2 WMMA tiles), K staged in LDS in
// chunks of 32:
//   ldsA[m][k] row-major      (async global->LDS b128 when available)
//   ldsB[n][k] TRANSPOSED     (b128 global load + ds_store scatter)
// Inner loop per K-step of 4: 4x ds_load_b64 + 4x v_wmma_f32_16x16x4_f32.
// Fragment layouts per CDNA5 ISA 7.12.2 (wave32):
//   A 16x4 : lane m = lane&15, K pair = 2*(lane>>4) + {reg0,reg1}
//   B 4x16 : lane n = lane&15, same K striping
//   C 16x16: vgpr r -> M = r + 8*(lane>>4), N = lane&15
// MODE 0: store D.   MODE 1: accumulate sum|Op - D| into *acc (wave reduce).
template <int MODE>
__global__ void k_gemm(const float* __restrict__ A, const float* __restrict__ B,
                       float* __restrict__ Cout, const float* __restrict__ Op,
                       float* __restrict__ acc) {
  __shared__ float ldsA[32 * BPAD];
  __shared__ float ldsB[32 * BPAD];

  const int ch = blockIdx.z;
  const float* a = A + (size_t)ch * CH_STRIDE;
  const float* b = B + (size_t)ch * CH_STRIDE;
  const int m0 = blockIdx.x * 32;
  const int n0 = blockIdx.y * 32;
  const int lane = threadIdx.x & 31;
  const int half = lane >> 4;
  const int l16 = lane & 15;
  // staging decomposition: 32 lanes cover 4 rows x (8 lanes * 4 floats)
  const int srow = lane >> 3;        // 0..3
  const int scol = (lane & 7) * 4;   // 0,4,...,28

  v8f c00 = {}, c01 = {}, c10 = {}, c11 = {};

  for (int kc = 0; kc < LDIM; kc += KB) {
    // ---- stage A chunk (rows m0..m0+31, k kc..kc+31), row-major ----------
#if HAVE_ASYNC_LDS
    #pragma unroll
    for (int i = 0; i < 8; ++i) {
      int r = srow + 4 * i;
      const float* g = a + (size_t)(m0 + r) * LDIM + kc + scol;
      __builtin_amdgcn_global_load_async_to_lds_b128(
          (gptr_v4i)g,
          (lptr_v4i)&ldsA[r * BPAD + scol],
          0, 0);
    }
#else
    #pragma unroll
    for (int i = 0; i < 8; ++i) {
      int r = srow + 4 * i;
      float4 v = *(const float4*)(a + (size_t)(m0 + r) * LDIM + kc + scol);
      *(float4*)&ldsA[r * BPAD + scol] = v;
    }
#endif
    // ---- stage B chunk (k kc..kc+31, cols n0..n0+31), TRANSPOSED ---------
    #pragma unroll
    for (int i = 0; i < 8; ++i) {
      int r = srow + 4 * i;  // k-row within chunk
      float4 v = *(const float4*)(b + (size_t)(kc + r) * LDIM + n0 + scol);
      ldsB[(scol + 0) * BPAD + r] = v.x;
      ldsB[(scol + 1) * BPAD + r] = v.y;
      ldsB[(scol + 2) * BPAD + r] = v.z;
      ldsB[(scol + 3) * BPAD + r] = v.w;
    }
    wait_async_lds();
    asm volatile("" ::: "memory");  // keep fragment reads after staging

    // ---- 8 K-steps of 4 --------------------------------------------------
    #pragma unroll
    for (int k2 = 0; k2 < KB; k2 += 4) {
      int kk = k2 + 2 * half;
      v2f a0 = *(const v2f*)&ldsA[l16 * BPAD + kk];
      v2f a1 = *(const v2f*)&ldsA[(l16 + 16) * BPAD + kk];
      v2f b0 = *(const v2f*)&ldsB[l16 * BPAD + kk];
      v2f b1 = *(const v2f*)&ldsB[(l16 + 16) * BPAD + kk];
      c00 = __builtin_amdgcn_wmma_f32_16x16x4_f32(false, a0, false, b0, (short)0, c00, false, false);
      c01 = __builtin_amdgcn_wmma_f32_16x16x4_f32(false, a0, false, b1, (short)0, c01, false, false);
      c10 = __builtin_amdgcn_wmma_f32_16x16x4_f32(false, a1, false, b0, (short)0, c10, false, false);
      c11 = __builtin_amdgcn_wmma_f32_16x16x4_f32(false, a1, false, b1, (short)0, c11, false, false);
    }
    asm volatile("" ::: "memory");  // keep next staging after fragment reads
  }

  if (MODE == 0) {
    float* cc = Cout + (size_t)ch * CH_STRIDE;
    #pragma unroll
    for (int r = 0; r < 8; ++r) {
      int row = r + half * 8;
      cc[(size_t)(m0 + row) * LDIM + n0 + l16]           = c00[r];
      cc[(size_t)(m0 + row) * LDIM + n0 + 16 + l16]      = c01[r];
      cc[(size_t)(m0 + 16 + row) * LDIM + n0 + l16]      = c10[r];
      cc[(size_t)(m0 + 16 + row) * LDIM + n0 + 16 + l16] = c11[r];
    }
  } else {
    const float* oc = Op + (size_t)ch * CH_STRIDE;
    float s = 0.f;
    #pragma unroll
    for (int r = 0; r < 8; ++r) {
      int row = r + half * 8;
      s += fabsf(oc[(size_t)(m0 + row) * LDIM + n0 + l16]           - c00[r]);
      s += fabsf(oc[(size_t)(m0 + row) * LDIM + n0 + 16 + l16]      - c01[r]);
      s += fabsf(oc[(size_t)(m0 + 16 + row) * LDIM + n0 + l16]      - c10[r]);
      s += fabsf(oc[(size_t)(m0 + 16 + row) * LDIM + n0 + 16 + l16] - c11[r]);
    }
    #pragma unroll
    for (int off = 16; off > 0; off >>= 1) s += __shfl_down(s, off, 32);
    if (lane == 0) atomicAdd(acc, s);
  }
}

// -------------------- rectification: column mean over H --------------------
__global__ void k_colmean(const float* __restrict__ x2, const float* __restrict__ xin,
                          float* __restrict__ cm) {
  int w = threadIdx.x;
  int ch = blockIdx.x;
  if (w >= VALIDD) return;
  const float* a = x2 + (size_t)ch * CH_STRIDE;
  const float* b = xin + (size_t)ch * CH_STRIDE;
  float s = 0.f;
  for (int h = 0; h < VALIDD; ++h) s += a[h * LDIM + w] - b[h * LDIM + w];
  cm[ch * LDIM + w] = s * (1.f / (float)VALIDD);
}

// x_out = x_in + relu((x2 - x_in) - colmean)   (x_out may alias x2)
__global__ void k_rect(const float* __restrict__ xin, const float* __restrict__ x2,
                       const float* __restrict__ cm, float* __restrict__ xout) {
  int idx = blockIdx.x * blockDim.x + threadIdx.x;
  if (idx >= CH_STRIDE) return;
  int ch = blockIdx.y;
  int h = idx / LDIM, w = idx % LDIM;
  size_t o = (size_t)ch * CH_STRIDE + idx;
  if (h >= VALIDD || w >= VALIDD) { xout[o] = 0.f; return; }
  float xi = xin[o];
  float off = x2[o] - xi;
  xout[o] = xi + fmaxf(off - cm[ch * LDIM + w], 0.f);
}

// ------------------ channel-axis L2 normalize (in place) -------------------
__global__ void k_norm(float* __restrict__ X) {
  int idx = blockIdx.x * blockDim.x + threadIdx.x;
  if (idx >= CH_STRIDE) return;
  float s = 0.f;
  for (int c = 0; c < CHANS; ++c) {
    float v = X[(size_t)c * CH_STRIDE + idx];
    s += v * v;
  }
  float inv = 1.f / fmaxf(sqrtf(s), 1e-12f);
  for (int c = 0; c < CHANS; ++c) X[(size_t)c * CH_STRIDE + idx] *= inv;
}

// ------------------------------- finalize ----------------------------------
__global__ void k_final(const float* __restrict__ acc, const float* __restrict__ base,
                        float* __restrict__ out) {
  out[0] = fabsf(acc[0] - acc[1]) + base[0];
}

// ---------------------------------------------------------------------------
extern "C" void kernel_launch(void* const* d_in, const int* in_sizes, int n_in,
                              void* d_out, int out_size, void* d_ws, size_t ws_size,
                              hipStream_t stream) {
  const float* logits = (const float*)d_in[0];   // (1,6,64,64) f32
  const float* base   = (const float*)d_in[1];   // (1,) f32
  float* out = (float*)d_out;

  float* ws = (float*)d_ws;
  const size_t SZ = (size_t)CHANS * CH_STRIDE;   // ~19.9 MB each
  float* fcp = ws;            // pad1(fc)
  float* fcn = ws + SZ;       // pad1(-fc)
  float* P   = ws + 2 * SZ;   // pooled
  float* X   = ws + 3 * SZ;   // iter buffer
  float* Y   = ws + 4 * SZ;   // iter buffer
  float* cm  = ws + 5 * SZ;   // 60*288 column means
  float* acc = cm + CHANS * LDIM;  // [0]=n_open, [1]=n_dil

  dim3 bpix(256);
  dim3 gpix(CH_STRIDE / 256);              // 324
  dim3 gpixc(CH_STRIDE / 256, CHANS);      // 324 x 60
  dim3 ggemm(LDIM / 32, LDIM / 32, CHANS); // 9 x 9 x 60, one wave each
  dim3 bgemm(32);

  k_init<<<1, 1, 0, stream>>>(acc);
  k_prep<<<gpix, bpix, 0, stream>>>(logits, fcp, fcn);

  // ---- opening: x = matmul(maxpool3(x), fcp), twice -----------------------
  k_pool<1><<<gpixc, bpix, 0, stream>>>(fcp, P);
  k_gemm<0><<<ggemm, bgemm, 0, stream>>>(P, fcp, X, nullptr, nullptr);
  k_pool<1><<<gpixc, bpix, 0, stream>>>(X, P);
  k_gemm<0><<<ggemm, bgemm, 0, stream>>>(P, fcp, Y, nullptr, nullptr);
  k_norm<<<gpix, bpix, 0, stream>>>(Y);
  // n_open = sum|fcn - matmul(norm(Y), fcn)|
  k_gemm<1><<<ggemm, bgemm, 0, stream>>>(Y, fcn, nullptr, fcn, acc);

  // ---- selective dilation: x2 = matmul(avgpool3(x), fcp); rectify; twice --
  k_pool<0><<<gpixc, bpix, 0, stream>>>(fcp, P);
  k_gemm<0><<<ggemm, bgemm, 0, stream>>>(P, fcp, X, nullptr, nullptr);
  k_colmean<<<CHANS, LDIM, 0, stream>>>(X, fcp, cm);
  k_rect<<<gpixc, bpix, 0, stream>>>(fcp, X, cm, X);
  k_pool<0><<<gpixc, bpix, 0, stream>>>(X, P);
  k_gemm<0><<<ggemm, bgemm, 0, stream>>>(P, fcp, Y, nullptr, nullptr);
  k_colmean<<<CHANS, LDIM, 0, stream>>>(Y, X, cm);
  k_rect<<<gpixc, bpix, 0, stream>>>(X, Y, cm, Y);
  k_norm<<<gpix, bpix, 0, stream>>>(Y);
  // n_dil = sum|fcp - matmul(norm(Y), fcp)|
  k_gemm<1><<<ggemm, bgemm, 0, stream>>>(Y, fcp, nullptr, fcp, acc + 1);

  k_final<<<1, 1, 0, stream>>>(acc, base, out);
}